// LFA_self_1022202216920
// MI455X (gfx1250) — compile-verified
//
#include <hip/hip_runtime.h>
#include <math.h>

// MI455X / gfx1250, wave32. Fully fused pointwise-proj + MHA + out-proj.
// One workgroup (256 threads = 8 waves) per (batch, head).
// fp32 WMMA (V_WMMA_F32_16X16X4_F32) for QK^T and P*V.

typedef float v2f __attribute__((ext_vector_type(2)));
typedef float v8f __attribute__((ext_vector_type(8)));

#define BB   32            // batch
#define D1   49            // tokens (SAIs)
#define SN   32            // sqrt_n
#define NF   16            // filters / channels
#define HH   16            // heads
#define CPAD 68            // LDS row stride for 64-wide chunk buffers (68%64==4 -> conflict free)
#define SPAD 68            // LDS row stride for score matrix

// Stage one depth-chunk (4 sites = 64 depth) of a projected tensor into LDS.
// dst[r][sl*16+f] = bias[f] + sum_c x[b, r, hs, ws, c] * W[c][f]   (rows >= 49 -> 0)
__device__ __forceinline__ void stage_proj(const float* __restrict__ x,
                                           const float* __restrict__ Wl,
                                           const float* __restrict__ Bl,
                                           float* __restrict__ dst,
                                           int b, int h, int ch, int tid)
{
    const int r  = tid >> 2;        // row 0..63
    const int sl = tid & 3;         // site within chunk
    const int s  = ch * 4 + sl;     // site within head 0..63
    float y[16];
    if (r < D1) {
        const int hs = 2 * h + (s >> 5);
        const int ws = s & 31;
        const float* xp = x + ((((size_t)b * D1 + r) * SN + hs) * SN + ws) * NF;
#pragma unroll
        for (int f = 0; f < 16; ++f) y[f] = Bl[f];
#pragma unroll
        for (int c = 0; c < 16; ++c) {
            const float xc = xp[c];
#pragma unroll
            for (int f = 0; f < 16; ++f) y[f] += xc * Wl[c * 16 + f];
        }
    } else {
#pragma unroll
        for (int f = 0; f < 16; ++f) y[f] = 0.0f;
    }
#pragma unroll
    for (int f = 0; f < 16; ++f) dst[r * CPAD + sl * 16 + f] = y[f];
}

__global__ __launch_bounds__(256) void lfa_fused_attn(
    const float* __restrict__ xv, const float* __restrict__ xk, const float* __restrict__ xq,
    const float* __restrict__ Wq, const float* __restrict__ bq,
    const float* __restrict__ Wk, const float* __restrict__ bk,
    const float* __restrict__ Wv, const float* __restrict__ bv,
    const float* __restrict__ Wo, const float* __restrict__ bo,
    float* __restrict__ out, float* __restrict__ attn)
{
    __shared__ float Qs[64 * CPAD];   // Q chunk (pass A) / ctx chunk (pass B)
    __shared__ float Ks[64 * CPAD];   // K chunk (pass A) / V  chunk (pass B)
    __shared__ float Ss[64 * SPAD];   // scores -> probabilities (64x64, 49x49 valid)
    __shared__ float Wl[4 * 256];     // Wq | Wk | Wv | Wo
    __shared__ float Bl[4 * 16];      // bq | bk | bv | bo

    const int tid  = threadIdx.x;
    const int h    = blockIdx.x;
    const int b    = blockIdx.y;
    const int lane = tid & 31;
    const int w    = tid >> 5;        // wave id 0..7
    const int lr   = lane & 15;
    const int hi   = lane >> 4;       // 0: K0/K1 half, 1: K2/K3 half

    // --- load weights/biases into LDS ---
    Wl[tid]        = Wq[tid];
    Wl[256 + tid]  = Wk[tid];
    Wl[512 + tid]  = Wv[tid];
    Wl[768 + tid]  = Wo[tid];
    if (tid < 16) {
        Bl[tid]      = bq[tid];
        Bl[16 + tid] = bk[tid];
        Bl[32 + tid] = bv[tid];
        Bl[48 + tid] = bo[tid];
    }
    __syncthreads();

    // ================= Pass A: S = Qh @ Kh^T  (64x64 padded, depth 1024) =========
    // 16 output tiles (4m x 4n); wave w owns tiles w and w+8.
    const int m0 = w >> 2, n0 = w & 3;
    const int m1 = m0 + 2;                       // tile w+8 -> (m0+2, n0)
    v8f s0 = {}; v8f s1 = {};

    for (int ch = 0; ch < 16; ++ch) {            // 16 chunks x 64 depth = 1024
        stage_proj(xq, &Wl[0],   &Bl[0],  Qs, b, h, ch, tid);
        stage_proj(xk, &Wl[256], &Bl[16], Ks, b, h, ch, tid);
        __syncthreads();

        const float* qrow0 = &Qs[(m0 * 16 + lr) * CPAD];
        const float* qrow1 = &Qs[(m1 * 16 + lr) * CPAD];
        const float* krow  = &Ks[(n0 * 16 + lr) * CPAD];
#pragma unroll 4
        for (int kk = 0; kk < 64; kk += 4) {
            const int ko = kk + 2 * hi;
            v2f a0 = *(const v2f*)&qrow0[ko];
            v2f bv2 = *(const v2f*)&krow[ko];
            s0 = __builtin_amdgcn_wmma_f32_16x16x4_f32(false, a0, false, bv2,
                                                       (short)0, s0, false, false);
            v2f a1 = *(const v2f*)&qrow1[ko];
            s1 = __builtin_amdgcn_wmma_f32_16x16x4_f32(false, a1, false, bv2,
                                                       (short)0, s1, false, false);
        }
        __syncthreads();
    }
    // store score tiles: C VGPR r holds M = r (lanes 0-15) / r+8 (lanes 16-31), N = lr
#pragma unroll
    for (int r = 0; r < 8; ++r) {
        Ss[(m0 * 16 + r + hi * 8) * SPAD + n0 * 16 + lr] = s0[r];
        Ss[(m1 * 16 + r + hi * 8) * SPAD + n0 * 16 + lr] = s1[r];
    }
    __syncthreads();

    // ================= Softmax over valid 49 cols, write attn ====================
    if (tid < 64) {
        const int r = tid;
        float* srow = &Ss[r * SPAD];
        if (r < D1) {
            const float scale = 0.03125f;        // 1/sqrt(1024)
            float mx = -3.402823e38f;
            for (int c = 0; c < D1; ++c) {
                float vsc = srow[c] * scale;
                srow[c] = vsc;
                mx = fmaxf(mx, vsc);
            }
            float sum = 0.0f;
            for (int c = 0; c < D1; ++c) {
                float e = __expf(srow[c] - mx);
                srow[c] = e;
                sum += e;
            }
            const float inv = 1.0f / sum;
            float* arow = &attn[(((size_t)b * HH + h) * D1 + r) * D1];
            for (int c = 0; c < D1; ++c) {
                float pv = srow[c] * inv;
                srow[c] = pv;
                arow[c] = pv;
            }
            for (int c = D1; c < 64; ++c) srow[c] = 0.0f;   // kill padded K columns
        } else {
            for (int c = 0; c < 64; ++c) srow[c] = 0.0f;    // padded Q rows
        }
    }
    __syncthreads();

    // ================= Pass B: ctx = P @ Vh, then out = ctx @ Wo + bo ============
    for (int ch = 0; ch < 16; ++ch) {
        stage_proj(xv, &Wl[512], &Bl[32], Ks, b, h, ch, tid);
        __syncthreads();

        // ctx chunk is [64 rows x 64 depth]: 16 tiles, wave w owns tiles w and w+8.
#pragma unroll
        for (int ti = 0; ti < 2; ++ti) {
            const int t = w + ti * 8;
            const int m = t >> 2, n = t & 3;
            v8f acc = {};
            const float* prow = &Ss[(m * 16 + lr) * SPAD];
            const float* vcol = &Ks[n * 16 + lr];
#pragma unroll 4
            for (int kk = 0; kk < 64; kk += 4) {
                const int ko = kk + 2 * hi;
                v2f a = *(const v2f*)&prow[ko];
                v2f bv2;
                bv2.x = vcol[ko * CPAD];            // B[k][n] = V[k][n*16+lr]
                bv2.y = vcol[(ko + 1) * CPAD];
                acc = __builtin_amdgcn_wmma_f32_16x16x4_f32(false, a, false, bv2,
                                                            (short)0, acc, false, false);
            }
#pragma unroll
            for (int r = 0; r < 8; ++r)
                Qs[(m * 16 + r + hi * 8) * CPAD + n * 16 + lr] = acc[r];
        }
        __syncthreads();

        // Fused output projection: out[b, r, hs, ws, f] = bo[f] + ctx . Wo
        {
            const int r  = tid >> 2;
            const int sl = tid & 3;
            if (r < D1) {
                const int s  = ch * 4 + sl;
                const int hs = 2 * h + (s >> 5);
                const int ws = s & 31;
                float y[16];
#pragma unroll
                for (int f = 0; f < 16; ++f) y[f] = Bl[48 + f];
#pragma unroll
                for (int fc = 0; fc < 16; ++fc) {
                    const float xc = Qs[r * CPAD + sl * 16 + fc];
#pragma unroll
                    for (int f = 0; f < 16; ++f) y[f] += xc * Wl[768 + fc * 16 + f];
                }
                float* op = out + ((((size_t)b * D1 + r) * SN + hs) * SN + ws) * NF;
#pragma unroll
                for (int f = 0; f < 16; ++f) op[f] = y[f];
            }
        }
        __syncthreads();
    }
}

extern "C" void kernel_launch(void* const* d_in, const int* in_sizes, int n_in,
                              void* d_out, int out_size, void* d_ws, size_t ws_size,
                              hipStream_t stream) {
    (void)in_sizes; (void)n_in; (void)out_size; (void)d_ws; (void)ws_size;
    const float* v  = (const float*)d_in[0];
    const float* k  = (const float*)d_in[1];
    const float* q  = (const float*)d_in[2];
    const float* Wq = (const float*)d_in[3];
    const float* bq = (const float*)d_in[4];
    const float* Wk = (const float*)d_in[5];
    const float* bk = (const float*)d_in[6];
    const float* Wv = (const float*)d_in[7];
    const float* bv = (const float*)d_in[8];
    const float* Wo = (const float*)d_in[9];
    const float* bo = (const float*)d_in[10];

    float* out  = (float*)d_out;
    float* attn = out + (size_t)BB * D1 * SN * SN * NF;   // outputs concatenated flat

    dim3 grid(HH, BB);   // one workgroup per (head, batch): 512 workgroups
    lfa_fused_attn<<<grid, 256, 0, stream>>>(v, k, q, Wq, bq, Wk, bk, Wv, bv, Wo, bo,
                                             out, attn);
}